// RGCN_69793218560327
// MI455X (gfx1250) — compile-verified
//
#include <hip/hip_runtime.h>

typedef __attribute__((ext_vector_type(2))) float v2f;
typedef __attribute__((ext_vector_type(8))) float v8f;

// ---------------------------------------------------------------------------
// helpers
// ---------------------------------------------------------------------------
__device__ __forceinline__ void atomAddF32(float* p, float v) {
#if __has_builtin(__builtin_amdgcn_global_atomic_fadd_f32)
  typedef __attribute__((address_space(1))) float gfloat;
  (void)__builtin_amdgcn_global_atomic_fadd_f32((gfloat*)p, v);
#else
  (void)__hip_atomic_fetch_add(p, v, __ATOMIC_RELAXED, __HIP_MEMORY_SCOPE_AGENT);
#endif
}

__global__ void fill_zero(float* __restrict__ p, long n) {
  long i = (long)blockIdx.x * blockDim.x + threadIdx.x;
  if (i < n) p[i] = 0.0f;
}

// count degrees: src -> degOut, dst -> degIn  (flat over R*E, idx layout [r*E+e])
__global__ void degree_count(const int* __restrict__ src, const int* __restrict__ dst,
                             float* __restrict__ degOut, float* __restrict__ degIn,
                             int N, int E, long RE) {
  long i = (long)blockIdx.x * blockDim.x + threadIdx.x;
  if (i >= RE) return;
  long r = i / E;
  int s = src[i];
  int d = dst[i];
  atomAddF32(&degOut[r * (long)N + s], 1.0f);
  atomAddF32(&degIn [r * (long)N + d], 1.0f);
}

// deg -> rsqrt(max(deg,1)) in place
__global__ void rinv_transform(float* __restrict__ p, long n) {
  long i = (long)blockIdx.x * blockDim.x + threadIdx.x;
  if (i >= n) return;
  p[i] = rsqrtf(fmaxf(p[i], 1.0f));
}

// h = relu(h + sum_r b[r][f])
__global__ void bias_relu(float* __restrict__ h, const float* __restrict__ b,
                          long total, int F) {
  long i = (long)blockIdx.x * blockDim.x + threadIdx.x;
  if (i >= total) return;
  int f = (int)(i % F);
  float bs = b[f] + b[F + f] + b[2 * F + f] + b[3 * F + f];
  float v = h[i] + bs;
  h[i] = v > 0.0f ? v : 0.0f;
}

// o += sum_r b[r][f]
__global__ void bias_add(float* __restrict__ o, const float* __restrict__ b,
                         long total, int F) {
  long i = (long)blockIdx.x * blockDim.x + threadIdx.x;
  if (i >= total) return;
  int f = (int)(i % F);
  o[i] += b[f] + b[F + f] + b[2 * F + f] + b[3 * F + f];
}

// ---------------------------------------------------------------------------
// WMMA f32 GEMM:  Y[N x F] = X[N x K] * W[K x F]   (row-major, K mult of 32,
// F = COL_WAVES*16).  Block: 256 threads = 8 waves = COL_WAVES col-tiles x
// (8/COL_WAVES) row groups; each wave computes ROW_TILES stacked 16x16 tiles
// with V_WMMA_F32_16X16X4_F32.  A staged in LDS (stride 36: conflict-free);
// B fragments read from L2-resident weights.
// ---------------------------------------------------------------------------
template <int COL_WAVES, int ROW_TILES>
__global__ __launch_bounds__(256) void gemm_wmma_f32(
    const float* __restrict__ X, const float* __restrict__ W,
    float* __restrict__ Y, int N, int K, int F) {
  constexpr int ROW_GROUPS = 8 / COL_WAVES;
  constexpr int BLOCK_ROWS = ROW_GROUPS * ROW_TILES * 16;
  constexpr int STRIDE = 36;  // padded row stride (floats), even & conflict-free
  __shared__ float lds[BLOCK_ROWS * STRIDE];

  const int tid = threadIdx.x;
  const int wave = tid >> 5;
  const int lane = tid & 31;
  const int colWave = wave % COL_WAVES;
  const int rowGroup = wave / COL_WAVES;
  const int col0 = colWave * 16;
  const int rowBase = blockIdx.x * BLOCK_ROWS;
  const int laneHi = lane >> 4;   // 0 or 1
  const int laneLo = lane & 15;

  v8f acc[ROW_TILES] = {};

  for (int k0 = 0; k0 < K; k0 += 32) {
    // ---- cooperative stage of X[BLOCK_ROWS x 32] into LDS ----
#pragma unroll
    for (int v = 0; v < BLOCK_ROWS / 32; ++v) {
      int flat = (tid + v * 256) * 4;       // element index in the 2D chunk
      int row = flat >> 5;                  // /32
      int kk = flat & 31;
      int grow = rowBase + row;
      float4 val{0.0f, 0.0f, 0.0f, 0.0f};
      if (grow < N) val = *(const float4*)(X + (long)grow * K + k0 + kk);
      *(float4*)&lds[row * STRIDE + kk] = val;
    }
    __syncthreads();

    // ---- 8 x (16x16x4) WMMA steps over this K chunk ----
#pragma unroll
    for (int kk = 0; kk < 32; kk += 4) {
      // B fragment (4x16): vgpr j, lane l -> W[k0+kk + (l/16)*2 + j][col0 + l%16]
      const float* wp = W + (long)(k0 + kk + laneHi * 2) * F + col0 + laneLo;
      v2f bf;
      bf.x = wp[0];
      bf.y = wp[F];
#pragma unroll
      for (int t = 0; t < ROW_TILES; ++t) {
        // A fragment (16x4): vgpr j, lane l -> row l%16, K = kk + (l/16)*2 + j
        int arow = (rowGroup * ROW_TILES + t) * 16 + laneLo;
        v2f af = *(const v2f*)&lds[arow * STRIDE + kk + laneHi * 2];
        acc[t] = __builtin_amdgcn_wmma_f32_16x16x4_f32(
            /*neg_a=*/false, af, /*neg_b=*/false, bf,
            /*c_mod=*/(short)0, acc[t], /*reuse_a=*/false, /*reuse_b=*/false);
      }
    }
    __syncthreads();
  }

  // ---- store: VGPR i -> M = i + (lane>=16 ? 8 : 0), N = lane%16 ----
#pragma unroll
  for (int t = 0; t < ROW_TILES; ++t) {
#pragma unroll
    for (int i = 0; i < 8; ++i) {
      int grow = rowBase + (rowGroup * ROW_TILES + t) * 16 + i + laneHi * 8;
      if (grow < N) Y[(long)grow * F + col0 + laneLo] = acc[t][i];
    }
  }
}

// ---------------------------------------------------------------------------
// Edge scatter: one wave32 per edge.
// Out[d,:] += rinvOut[s] * rinvIn[d] * Yr[s,:]   (WF = 128 or 64 floats wide)
// Edge id forced into an SGPR so indices/degree factors become SMEM loads,
// keeping the vector pipe for the payload gather + f32 atomics.
// ---------------------------------------------------------------------------
template <int WF>
__global__ __launch_bounds__(256) void scatter_edges(
    const float* __restrict__ Yr, const int* __restrict__ src,
    const int* __restrict__ dst, const float* __restrict__ rinvOut,
    const float* __restrict__ rinvIn, float* __restrict__ Out, int E) {
  int e = __builtin_amdgcn_readfirstlane((int)blockIdx.x * 8 + (int)(threadIdx.x >> 5));
  int lane = threadIdx.x & 31;
  if (e >= E) return;
  int s = src[e];
  int d = dst[e];
  float coef = rinvOut[s] * rinvIn[d];
  if constexpr (WF == 128) {
    float4 v = *(const float4*)(Yr + (long)s * 128 + lane * 4);
    float* o = Out + (long)d * 128 + lane * 4;
    atomAddF32(o + 0, coef * v.x);
    atomAddF32(o + 1, coef * v.y);
    atomAddF32(o + 2, coef * v.z);
    atomAddF32(o + 3, coef * v.w);
  } else {
    float2 v = *(const float2*)(Yr + (long)s * 64 + lane * 2);
    float* o = Out + (long)d * 64 + lane * 2;
    atomAddF32(o + 0, coef * v.x);
    atomAddF32(o + 1, coef * v.y);
  }
}

// ---------------------------------------------------------------------------
// host launcher
// ---------------------------------------------------------------------------
extern "C" void kernel_launch(void* const* d_in, const int* in_sizes, int n_in,
                              void* d_out, int out_size, void* d_ws,
                              size_t ws_size, hipStream_t stream) {
  const float* x  = (const float*)d_in[0];
  const int*  src = (const int*)d_in[1];
  const int*  dst = (const int*)d_in[2];
  const float* W1 = (const float*)d_in[3];
  const float* b1 = (const float*)d_in[4];
  const float* W2 = (const float*)d_in[5];
  const float* b2 = (const float*)d_in[6];
  float* out = (float*)d_out;

  const int IN_F = 128, H_F = 128, OUT_F = 64, R = 4;
  const int N = in_sizes[0] / IN_F;
  const int E = in_sizes[1] / R;

  // workspace carve-out (~106 MB): degOut | degIn | h | y
  float* degOut = (float*)d_ws;                  // R*N
  float* degIn  = degOut + (long)R * N;          // R*N
  float* h      = degIn + (long)R * N;           // N*H_F
  float* y      = h + (long)N * H_F;             // N*H_F (reused per relation)

  auto cdiv = [](long a, long b) { return (int)((a + b - 1) / b); };

  // zero accumulators + degree counters
  fill_zero<<<cdiv(2L * R * N, 256), 256, 0, stream>>>(degOut, 2L * R * N);
  fill_zero<<<cdiv((long)N * H_F, 256), 256, 0, stream>>>(h, (long)N * H_F);
  fill_zero<<<cdiv((long)N * OUT_F, 256), 256, 0, stream>>>(out, (long)N * OUT_F);

  // degrees -> rsqrt(max(deg,1))
  degree_count<<<cdiv((long)R * E, 256), 256, 0, stream>>>(src, dst, degOut,
                                                           degIn, N, E,
                                                           (long)R * E);
  rinv_transform<<<cdiv(2L * R * N, 256), 256, 0, stream>>>(degOut, 2L * R * N);

  // ---- layer 1: per relation, y = x @ W1_r (WMMA), then edge scatter into h
  for (int r = 0; r < R; ++r) {
    gemm_wmma_f32<8, 4><<<cdiv(N, 64), 256, 0, stream>>>(
        x, W1 + (long)r * IN_F * H_F, y, N, IN_F, H_F);
    scatter_edges<128><<<cdiv(E, 8), 256, 0, stream>>>(
        y, src + (long)r * E, dst + (long)r * E, degOut + (long)r * N,
        degIn + (long)r * N, h, E);
  }
  bias_relu<<<cdiv((long)N * H_F, 256), 256, 0, stream>>>(h, b1,
                                                          (long)N * H_F, H_F);

  // ---- layer 2: per relation, y = h @ W2_r (width 64), scatter into out
  for (int r = 0; r < R; ++r) {
    gemm_wmma_f32<4, 4><<<cdiv(N, 128), 256, 0, stream>>>(
        h, W2 + (long)r * H_F * OUT_F, y, N, H_F, OUT_F);
    scatter_edges<64><<<cdiv(E, 8), 256, 0, stream>>>(
        y, src + (long)r * E, dst + (long)r * E, degOut + (long)r * N,
        degIn + (long)r * N, out, E);
  }
  bias_add<<<cdiv((long)N * OUT_F, 256), 256, 0, stream>>>(out, b2,
                                                           (long)N * OUT_F,
                                                           OUT_F);
}